// Model_14121852470213
// MI455X (gfx1250) — compile-verified
//
#include <hip/hip_runtime.h>

typedef __attribute__((ext_vector_type(2))) float v2f;
typedef __attribute__((ext_vector_type(4))) float v4f;
typedef __attribute__((ext_vector_type(8))) float v8f;

#define RAF_DT (1.0f / 24000.0f)

// One wave (32 lanes) integrates 16 batch sequences for the full T-step scan.
// lane = 16*h + m : m = batch-in-tile (also WMMA N column), h = half.
// Lane (h,m) owns RAF neurons d = 4c + 2h + i (c=0..3, i=0..1) -> this IS the
// fp32 4x16 B-fragment striping, so spike values feed V_WMMA_F32_16X16X4_F32
// directly. A = constant W_ext rows (classes 0..2 live, rest zero), K tiled as
// 5 chunks of 4 covering [16 raf spikes | dt_spk | ones(bias) | 0 | 0].
// D[M=class][N=batch] lands class r in VGPR r, batch in lane -> per-lane LIF.
__global__ __launch_bounds__(32) void raf_snn_scan(
    const float* __restrict__ x,        // [256, T]
    const float* __restrict__ omegas,   // [2, 16]
    const float* __restrict__ bsp,      // [2, 16]
    const float* __restrict__ rth,      // [16]
    const float* __restrict__ fcw,      // [3, 17]
    const float* __restrict__ fcb,      // [3]
    const float* __restrict__ lif_beta_p,
    const float* __restrict__ lif_th_p,
    float* __restrict__ out_spk,        // [T, 256, 3]
    float* __restrict__ out_mem,        // [T, 256, 3]
    int T) {
  const int lane = threadIdx.x;
  const int m = lane & 15;
  const int h = lane >> 4;
  const int bglob = blockIdx.x * 16 + m;
  const float beta = lif_beta_p[0];
  const float lifth = lif_th_p[0];
  const float h0 = (h == 0) ? 1.0f : 0.0f;

  // ---- per-lane RAF state: 8 neurons d = 4*(j>>1) + 2h + (j&1) ----
  float u[8], v[8], om2[8], dOm[8], b2[8], dB[8], th[8], sel[8];
#pragma unroll
  for (int j = 0; j < 8; ++j) {
    const int d = 4 * (j >> 1) + 2 * h + (j & 1);
    const float o1 = omegas[d], o2 = omegas[16 + d];
    const float c1 = bsp[d], c2 = bsp[16 + d];
    om2[j] = o2; dOm[j] = o1 - o2;
    b2[j] = c2;  dB[j] = c1 - c2;
    th[j] = rth[d];
    u[j] = 0.0f; v[j] = 0.0f;
    sel[j] = 1.0f;  // use_t1 = true
  }

  // ---- constant A fragments: A[M=m][K = 4c + 2h + i] of W_ext[16 x 20] ----
  // W_ext[m][k] = fc_w[m][k] for k<17, fc_b[m] at k==17, else 0 (m>=3 -> 0).
  v2f aW[5];
#pragma unroll
  for (int c = 0; c < 5; ++c) {
    float a0 = 0.0f, a1 = 0.0f;
    const int k0 = 4 * c + 2 * h, k1 = k0 + 1;
    if (m < 3) {
      a0 = (k0 < 17) ? fcw[m * 17 + k0] : ((k0 == 17) ? fcb[m] : 0.0f);
      a1 = (k1 < 17) ? fcw[m * 17 + k1] : ((k1 == 17) ? fcb[m] : 0.0f);
    }
    aW[c][0] = a0; aW[c][1] = a1;
  }

  float dtm = 0.0f;       // dual-threshold LIF membrane (per batch, h==0 live)
  v8f mem = {};           // D-layout LIF membrane: VGPR r = class r (r<3 live)
  const float* xp = x + (size_t)bglob * (size_t)T;
  const int TB3 = 0;  (void)TB3;

  for (int t0 = 0; t0 < T; t0 += 4) {
    const v4f xv = *(const v4f*)(xp + t0);
    const int pf = t0 + 256;
    __builtin_prefetch(xp + ((pf < T) ? pf : 0), 0, 3);  // global_prefetch
#pragma unroll
    for (int s = 0; s < 4; ++s) {
      const int t = t0 + s;
      const float xt = xv[s];
      const float curr = fminf(fmaxf(xt, -1.0f), 1.0f);

      // ---- resonate-and-fire bank with t1/t2 parameter switching ----
      v2f bf[5];
#pragma unroll
      for (int j = 0; j < 8; ++j) {
        const float om = fmaf(sel[j], dOm[j], om2[j]);
        const float bb = fmaf(sel[j], dB[j], b2[j]);
        const float un = fmaf(RAF_DT, fmaf(bb, u[j], fmaf(-om, v[j], curr)), u[j]);
        float vn = fmaf(RAF_DT, fmaf(om, u[j], bb * v[j]), v[j]);
        const float sp = (vn > th[j]) ? 1.0f : 0.0f;   // spike(vn - th)
        vn = fmaf(-sp, th[j], vn);                      // subtract reset
        sel[j] = fabsf(sel[j] - sp);                    // toggle param set
        u[j] = un; v[j] = vn;
        bf[j >> 1][j & 1] = sp;                         // B fragment: spikes^T
      }

      // ---- dual-threshold LIF on raw input (beta=0.9 const) ----
      dtm = fmaf(0.9f, dtm, xt);
      const float pos = (dtm > 1.0f) ? 1.0f : 0.0f;
      const float neg = (dtm < -1.0f) ? 1.0f : 0.0f;
      const float dspk = pos - neg;
      dtm = dtm - pos + neg;                            // subtract both resets
      bf[4][0] = h0 * dspk;                             // K=16 row: dt_spk
      bf[4][1] = h0;                                    // K=17 row: ones (bias)

      // ---- readout: mem = beta*mem + W_ext @ combined^T via 5x fp32 WMMA ----
      v8f acc;
#pragma unroll
      for (int r = 0; r < 8; ++r) acc[r] = beta * mem[r];
      acc = __builtin_amdgcn_wmma_f32_16x16x4_f32(false, aW[0], false, bf[0],
                                                  (short)0, acc, false, false);
      acc = __builtin_amdgcn_wmma_f32_16x16x4_f32(false, aW[1], false, bf[1],
                                                  (short)0, acc, false, false);
      acc = __builtin_amdgcn_wmma_f32_16x16x4_f32(false, aW[2], false, bf[2],
                                                  (short)0, acc, false, false);
      acc = __builtin_amdgcn_wmma_f32_16x16x4_f32(false, aW[3], false, bf[3],
                                                  (short)0, acc, false, false);
      acc = __builtin_amdgcn_wmma_f32_16x16x4_f32(false, aW[4], false, bf[4],
                                                  (short)0, acc, false, false);

      // ---- output LIF spike + reset (classes 0..2 live in VGPRs 0..2) ----
      const float s0 = (acc[0] > lifth) ? 1.0f : 0.0f;
      const float s1 = (acc[1] > lifth) ? 1.0f : 0.0f;
      const float s2 = (acc[2] > lifth) ? 1.0f : 0.0f;
      acc[0] = fmaf(-s0, lifth, acc[0]);
      acc[1] = fmaf(-s1, lifth, acc[1]);
      acc[2] = fmaf(-s2, lifth, acc[2]);
      mem = acc;

      if (h == 0) {  // lanes 0..15 own per-batch results
        const size_t base = ((size_t)t * 256 + (size_t)bglob) * 3;
        out_spk[base + 0] = s0;
        out_spk[base + 1] = s1;
        out_spk[base + 2] = s2;
        out_mem[base + 0] = acc[0];
        out_mem[base + 1] = acc[1];
        out_mem[base + 2] = acc[2];
      }
    }
  }
}

extern "C" void kernel_launch(void* const* d_in, const int* in_sizes, int n_in,
                              void* d_out, int out_size, void* d_ws,
                              size_t ws_size, hipStream_t stream) {
  const float* x      = (const float*)d_in[0];  // [256, T]
  const float* omegas = (const float*)d_in[1];  // [2, 16]
  const float* bsp    = (const float*)d_in[2];  // [2, 16]
  const float* rth    = (const float*)d_in[3];  // [16]
  const float* fcw    = (const float*)d_in[4];  // [3, 17]
  const float* fcb    = (const float*)d_in[5];  // [3]
  const float* lbeta  = (const float*)d_in[6];  // scalar
  const float* lth    = (const float*)d_in[7];  // scalar

  const int T = in_sizes[0] / 256;
  float* out_spk = (float*)d_out;
  float* out_mem = out_spk + (size_t)T * 256 * 3;

  hipLaunchKernelGGL(raf_snn_scan, dim3(16), dim3(32), 0, stream,
                     x, omegas, bsp, rth, fcw, fcb, lbeta, lth,
                     out_spk, out_mem, T);
}